// adapt_smoothing_11742440587615
// MI455X (gfx1250) — compile-verified
//
#include <hip/hip_runtime.h>

typedef __attribute__((ext_vector_type(2))) float v2f;
typedef __attribute__((ext_vector_type(8))) float v8f;

#define K_TAPS 25
#define NCHUNK 10              // K padded to 40 = 10 steps of K=4
#define TS     64              // s-rows per block (4 waves x 16)
#define TF     16              // f-cols per block
#define SLAB_ROWS (TS + K_TAPS - 1)   // 88
#define BLOCK_THREADS 128

// Problem shape is fixed by the reference; bake it in so row strides fold into
// the 24-bit instruction offsets and base addressing becomes shifts.
#define DIM_B 32
#define DIM_S 4096
#define DIM_F 512

__global__ __launch_bounds__(BLOCK_THREADS)
void causal_win_wmma(const float* __restrict__ x,
                     const float* __restrict__ w,
                     float* __restrict__ out) {
    __shared__ float slab[SLAB_ROWS * TF];

    const int f0      = blockIdx.x * TF;
    const int s_block = blockIdx.y * TS;
    const int b       = blockIdx.z;

    const int t    = threadIdx.x;
    const int lane = t & 31;
    const int wid  = t >> 5;        // wave id 0..3 (wave32)
    const int hf   = lane >> 4;     // half-wave select (A/B K-pair, C/D row+8)
    const int col  = lane & 15;

    // ---- cooperative slab load: rows [s_block-24, s_block+63] clamped, 16 f-cols ----
    // float4 granularity: 88*16/4 = 352 vec4 elements
    const int n4 = SLAB_ROWS * TF / 4;
    for (int idx4 = t; idx4 < n4; idx4 += BLOCK_THREADS) {
        int row = idx4 >> 2;        // slab row 0..87
        int c4  = idx4 & 3;         // which float4 within the 16-wide row
        int s   = s_block + row - (K_TAPS - 1);
        if (s < 0) s = 0;           // causal edge replication
        const float4* src = reinterpret_cast<const float4*>(
            x + ((size_t)b * DIM_S + s) * DIM_F + f0 + 4 * c4);
        reinterpret_cast<float4*>(slab)[idx4] = *src;   // 4*idx4 == row*16 + 4*c4
    }

    // ---- banded weights matrix A-fragments: W[i,m] = w[m-i] if 0<=m-i<25 else 0 ----
    // A 16x4 f32 layout: lanes 0-15 hold row i=col at K={4k,4k+1}; lanes 16-31 at K={4k+2,4k+3}
    v2f aw[NCHUNK];
#pragma unroll
    for (int k = 0; k < NCHUNK; ++k) {
        int m0 = 4 * k + 2 * hf;
        int d0 = m0 - col;
        int d1 = d0 + 1;
        aw[k].x = (d0 >= 0 && d0 < K_TAPS) ? w[d0] : 0.0f;
        aw[k].y = (d1 >= 0 && d1 < K_TAPS) ? w[d1] : 0.0f;
    }

    __syncthreads();

    // ---- 10 chained fp32 WMMAs: D = W(16x40) * X(40x16) ----
    v8f c = {};
    const int rbase = 16 * wid + 2 * hf;   // wave's slab-row origin for its K-pair
#pragma unroll
    for (int k = 0; k < NCHUNK; ++k) {
        int r0 = rbase + 4 * k;
        v2f bx;
        bx.x = slab[r0 * TF + col];         // B 4x16: K-row r0,   N=col
        bx.y = slab[(r0 + 1) * TF + col];   //         B 4x16: K-row r0+1, N=col
        c = __builtin_amdgcn_wmma_f32_16x16x4_f32(
                /*neg_a=*/false, aw[k], /*neg_b=*/false, bx,
                /*c_mod=*/(short)0, c, /*reuse_a=*/false, /*reuse_b=*/false);
    }

    // ---- store: one base pointer; constant row stride r*512 floats = r*2048 B
    //      folds into the 24-bit instruction offset -> 8 stores, one vaddr ----
    const int srow0 = s_block + 16 * wid + (hf ? 8 : 0);   // C/D VGPR r -> row r (+8 upper half)
    float* __restrict__ dst = out + ((size_t)b * DIM_S + srow0) * DIM_F + f0 + col;
#pragma unroll
    for (int r = 0; r < 8; ++r) {
        dst[r * DIM_F] = c[r];
    }
}

extern "C" void kernel_launch(void* const* d_in, const int* in_sizes, int n_in,
                              void* d_out, int out_size, void* d_ws, size_t ws_size,
                              hipStream_t stream) {
    const float* x = (const float*)d_in[0];   // [B, S, F] f32
    const float* w = (const float*)d_in[1];   // [K] f32
    float* out = (float*)d_out;               // [B, S, F] f32

    dim3 grid(DIM_F / TF, DIM_S / TS, DIM_B); // (32, 64, 32) = 65536 blocks
    causal_win_wmma<<<grid, BLOCK_THREADS, 0, stream>>>(x, w, out);
}